// Att_trans_cat_inte_42984032698914
// MI455X (gfx1250) — compile-verified
//
#include <hip/hip_runtime.h>

#define EM_D 64
#define SEQ_N 512
#define NEG_SLOPE 0.01f

typedef __attribute__((ext_vector_type(2))) float v2f;
typedef __attribute__((ext_vector_type(4))) float v4f;
typedef __attribute__((ext_vector_type(8))) float v8f;

// Workspace float layout:
// [0:64)   qv      [64:128) kv      [128:192) cv     [192] Kc
// [256:2304)  rq[b*512+n]       [2304:4352) rk[b*512+n]

__global__ void prep_vecs(const float* __restrict__ W_w, const float* __restrict__ W_b,
                          const float* __restrict__ a_w, const float* __restrict__ a_b,
                          float* __restrict__ ws) {
  int d = threadIdx.x;  // 64 threads
  float q = 0.f, k = 0.f, c = 0.f;
  for (int e = 0; e < EM_D; ++e) {
    float w = W_w[e * EM_D + d];
    q += w * a_w[e];
    k += w * a_w[EM_D + e];
    c += w * a_w[2 * EM_D + e];
  }
  ws[d] = q;
  ws[EM_D + d] = k;
  ws[2 * EM_D + d] = c;
  if (d == 0) {
    float kc = 0.f;
    for (int e = 0; e < EM_D; ++e)
      kc += W_b[e] * (a_w[e] + a_w[EM_D + e] + a_w[2 * EM_D + e]);
    ws[192] = kc + a_b[0];
  }
}

__global__ void prep_rows(const float* __restrict__ emb, float* __restrict__ ws) {
  int row = blockIdx.x * blockDim.x + threadIdx.x;  // 0..2047  (= b*512+n)
  const float* e = emb + (size_t)row * EM_D;
  float rq = 0.f, rk = 0.f;
  for (int d = 0; d < EM_D; ++d) {
    float v = e[d];
    rq += v * ws[d];
    rk += v * ws[EM_D + d];
  }
  ws[256 + row] = rq;
  ws[256 + 4 * SEQ_N + row] = rk;
}

__global__ __launch_bounds__(256) void attn_value_kernel(
    const float* __restrict__ emb, const float* __restrict__ ws, float* __restrict__ out) {
  const int b    = blockIdx.x >> 5;        // 4 batches
  const int i0   = (blockIdx.x & 31) << 4; // 32 i-tiles of 16
  const int tid  = threadIdx.x;
  const int wave = tid >> 5;
  const int lane = tid & 31;
  const int m16  = lane & 15;   // row-in-tile (A) / col-in-tile (B)
  const int khalf = lane >> 4;  // selects K pair {0,1} vs {2,3}

  __shared__ __align__(16) float emb_i[16][EM_D];   // 4 KB
  __shared__ float s_buf[16][SEQ_N + 4];            // ~32 KB, padded vs bank conflicts
  __shared__ float rk_row[SEQ_N];                   // 2 KB
  __shared__ float cv_s[EM_D];
  __shared__ float rq_t[16];

  const float* embB = emb + (size_t)b * SEQ_N * EM_D;

  // ---- phase 0: stage tile-invariant data in LDS ----
  for (int t = tid; t < 16 * EM_D; t += 256)
    emb_i[t >> 6][t & 63] = embB[(size_t)(i0 + (t >> 6)) * EM_D + (t & 63)];
  for (int t = tid; t < SEQ_N; t += 256)
    rk_row[t] = ws[256 + 4 * SEQ_N + b * SEQ_N + t];
  if (tid < EM_D) cv_s[tid] = ws[2 * EM_D + tid];
  if (tid < 16)   rq_t[tid] = ws[256 + b * SEQ_N + i0 + tid];
  const float Kc = ws[192];
  __syncthreads();

  // ---- phase 1: scores via fp32 WMMA: G = (emb_i .* cv) @ emb^T ----
  // Wave-uniform loop => EXEC all ones for every v_wmma.
  for (int jt = wave; jt < 32; jt += 8) {
    const int j0 = jt << 4;
    v8f acc = {0.f, 0.f, 0.f, 0.f, 0.f, 0.f, 0.f, 0.f};
    const float* bp0 = embB + (size_t)(j0 + m16) * EM_D;
    for (int k0 = 0; k0 < EM_D; k0 += 4) {
      const int col = k0 + (khalf << 1);
      v2f av, bv;
      // A 16x4 layout: VGPR0 = A[M][2*khalf], VGPR1 = A[M][2*khalf+1], M = lane%16
      av[0] = emb_i[m16][col]     * cv_s[col];
      av[1] = emb_i[m16][col + 1] * cv_s[col + 1];
      // B 4x16 layout (row striped across lanes): VGPRv = B[v + 2*khalf][N], N = lane%16
      bv[0] = bp0[col];
      bv[1] = bp0[col + 1];
      acc = __builtin_amdgcn_wmma_f32_16x16x4_f32(false, av, false, bv,
                                                  (short)0, acc, false, false);
    }
#pragma unroll
    for (int v = 0; v < 8; ++v) {
      const int m = v + (khalf << 3);  // C/D layout: M = v + 8*(lane/16), N = lane%16
      float s = acc[v] + rq_t[m] + rk_row[j0 + m16] + Kc;
      s = (s >= 0.f) ? s : NEG_SLOPE * s;  // leaky relu
      s_buf[m][j0 + m16] = s;
    }
  }
  __syncthreads();

  // ---- phase 2: softmax over j (wave-level shuffle reduction), write alphas ----
  for (int r = wave * 2; r < wave * 2 + 2; ++r) {
    float vals[16];
    float mx = -3.4e38f;
#pragma unroll
    for (int t = 0; t < 16; ++t) {
      vals[t] = s_buf[r][lane + (t << 5)];
      mx = fmaxf(mx, vals[t]);
    }
#pragma unroll
    for (int off = 16; off >= 1; off >>= 1) mx = fmaxf(mx, __shfl_xor(mx, off, 32));
    float sum = 0.f;
#pragma unroll
    for (int t = 0; t < 16; ++t) {
      vals[t] = __expf(vals[t] - mx);
      sum += vals[t];
    }
#pragma unroll
    for (int off = 16; off >= 1; off >>= 1) sum += __shfl_xor(sum, off, 32);
    const float inv = 1.0f / sum;
    float* arow = out + (size_t)(b * SEQ_N + i0 + r) * SEQ_N;
#pragma unroll
    for (int t = 0; t < 16; ++t) arow[lane + (t << 5)] = vals[t] * inv;
  }

  // ---- phase 3: stream value[b,i,j,:] = emb_i * emb_j (bandwidth-dominant) ----
  // lanes 0..15 cover d in float4 chunks, tid/16 covers j%16 -> 512B contiguous per wave store.
  const int d4   = (tid & 15) << 2;
  const int jsub = tid >> 4;
  float* vbase = out + (size_t)4 * SEQ_N * SEQ_N +
                 (size_t)(b * SEQ_N + i0) * SEQ_N * EM_D;
  for (int jb = 0; jb < 32; ++jb) {
    const int j = (jb << 4) + jsub;
    const v4f ej = *(const v4f*)(embB + (size_t)j * EM_D + d4);
#pragma unroll
    for (int m = 0; m < 16; ++m) {
      const v4f ei = *(const v4f*)(&emb_i[m][d4]);
      const v4f r = ei * ej;
      // streaming output (273 MB > L2): non-temporal store
      __builtin_nontemporal_store(r, (v4f*)(vbase + ((size_t)m * SEQ_N + j) * EM_D + d4));
    }
  }
}

extern "C" void kernel_launch(void* const* d_in, const int* in_sizes, int n_in,
                              void* d_out, int out_size, void* d_ws, size_t ws_size,
                              hipStream_t stream) {
  const float* emb = (const float*)d_in[0];
  const float* W_w = (const float*)d_in[1];
  const float* W_b = (const float*)d_in[2];
  const float* a_w = (const float*)d_in[3];
  const float* a_b = (const float*)d_in[4];
  float* out = (float*)d_out;
  float* ws  = (float*)d_ws;

  hipLaunchKernelGGL(prep_vecs, dim3(1), dim3(64), 0, stream, W_w, W_b, a_w, a_b, ws);
  hipLaunchKernelGGL(prep_rows, dim3(8), dim3(256), 0, stream, emb, ws);
  hipLaunchKernelGGL(attn_value_kernel, dim3(4 * (SEQ_N / 16)), dim3(256), 0, stream,
                     emb, ws, out);
}